// POM1bLoss_9732395893213
// MI455X (gfx1250) — compile-verified
//
#include <hip/hip_runtime.h>
#include <stdint.h>

#define NUM_CLASSES   7
#define ROWS_PER_TILE 256
#define TILE_BYTES    (ROWS_PER_TILE * NUM_CLASSES * 4)   // 7168 B per tile
#define BLOCK_THREADS 256
#define NUM_BLOCKS    2048

// Low 32 bits of a generic pointer into LDS == byte offset within the
// workgroup's LDS allocation (CDNA5 aperture rules, ISA 10.2).
__device__ __forceinline__ uint32_t lds_byte_off(const void* p) {
  return (uint32_t)(uintptr_t)p;
}

// Issue exactly TWO gfx1250 async b128 copies per lane (uniform per wave so
// ASYNCcnt advances by 2 per wave per tile; chunks 384..447 are covered twice
// -- benign duplicate writes of identical bytes).
__device__ __forceinline__ void issue_tile_copy(uint64_t gtile, uint32_t ltile,
                                                uint32_t c0, uint32_t c1) {
  uint32_t l0 = ltile + c0 * 16u;
  uint64_t g0 = gtile + (uint64_t)c0 * 16u;
  asm volatile("global_load_async_to_lds_b128 %0, %1, off"
               :: "v"(l0), "v"(g0) : "memory");
  uint32_t l1 = ltile + c1 * 16u;
  uint64_t g1 = gtile + (uint64_t)c1 * 16u;
  asm volatile("global_load_async_to_lds_b128 %0, %1, off"
               :: "v"(l1), "v"(g1) : "memory");
}

__global__ void pom1b_partial(const float* __restrict__ logits,
                              const int*  __restrict__ targets,
                              float* __restrict__ partials,
                              int numTiles) {
  __shared__ __align__(16) float lbuf[2][ROWS_PER_TILE * NUM_CLASSES]; // 2x7168B
  __shared__ float wsum[BLOCK_THREADS / 32];

  const int tid = threadIdx.x;
  // chunk ids: c0 in [0,256), c1 in [256,448) with tids 192..255 duplicating
  const uint32_t c0 = (uint32_t)tid;
  const uint32_t c1 = 256u + (uint32_t)(tid >= 192 ? tid - 64 : tid);
  const uint32_t lbase   = lds_byte_off(&lbuf[0][0]);
  const uint64_t gbase   = (uint64_t)(uintptr_t)logits;

  float acc = 0.0f;

  int tile = blockIdx.x;
  int buf  = 0;
  if (tile < numTiles)
    issue_tile_copy(gbase + (uint64_t)tile * TILE_BYTES, lbase, c0, c1);

  for (; tile < numTiles; tile += gridDim.x, buf ^= 1) {
    const int next = tile + gridDim.x;
    if (next < numTiles) {
      // prefetch next tile into the other buffer, then wait only for tile t
      issue_tile_copy(gbase + (uint64_t)next * TILE_BYTES,
                      lbase + (uint32_t)((buf ^ 1) * TILE_BYTES), c0, c1);
      asm volatile("s_wait_asynccnt 2" ::: "memory");
    } else {
      asm volatile("s_wait_asynccnt 0" ::: "memory");
    }
    __syncthreads();                       // all waves' tile-t copies visible

    // ---- one row per thread; stride-7 LDS reads are bank-conflict-free ----
    float l[NUM_CLASSES];
#pragma unroll
    for (int c = 0; c < NUM_CLASSES; ++c) l[c] = lbuf[buf][tid * NUM_CLASSES + c];

    const int row = tile * ROWS_PER_TILE + tid;
    const int k   = targets[row];                      // in [0, 6]

    float m = l[0];
#pragma unroll
    for (int c = 1; c < NUM_CLASSES; ++c) m = fmaxf(m, l[c]);
    float s = 0.0f;
#pragma unroll
    for (int c = 0; c < NUM_CLASSES; ++c)
      s += __builtin_amdgcn_exp2f((l[c] - m) * 1.4426950408889634f);
    const float lse = m + 0.69314718055994531f * __builtin_amdgcn_logf(s);

    // sum of logits over valid neighbors {k-1,k,k+1}; selects, no dyn indexing
    float sum_logit = 0.0f;
#pragma unroll
    for (int c = 0; c < NUM_CLASSES; ++c) {
      const int d = c - k;
      sum_logit += (d >= -1 && d <= 1) ? l[c] : 0.0f;
    }
    const float nvalid = 3.0f - (k == 0 ? 1.0f : 0.0f)
                              - (k == NUM_CLASSES - 1 ? 1.0f : 0.0f);
    acc += nvalid * lse - sum_logit;

    __syncthreads();   // buf fully consumed before tile t+2's issue reuses it
  }

  // ---- wave32 reduction, then cross-wave via LDS (deterministic) ----
#pragma unroll
  for (int off = 16; off > 0; off >>= 1)
    acc += __shfl_xor(acc, off, 32);
  if ((tid & 31) == 0) wsum[tid >> 5] = acc;
  __syncthreads();
  if (tid == 0) {
    float b = 0.0f;
#pragma unroll
    for (int w = 0; w < BLOCK_THREADS / 32; ++w) b += wsum[w];
    partials[blockIdx.x] = b;
  }
}

__global__ void pom1b_finalize(const float* __restrict__ partials, int n,
                               float* __restrict__ out, int totalRows) {
  __shared__ double red[256];
  const int tid = threadIdx.x;
  double s = 0.0;
  for (int i = tid; i < n; i += 256) s += (double)partials[i];
  red[tid] = s;
  __syncthreads();
  for (int stride = 128; stride > 0; stride >>= 1) {
    if (tid < stride) red[tid] += red[tid + stride];
    __syncthreads();
  }
  if (tid == 0) out[0] = (float)(red[0] / (double)totalRows);
}

extern "C" void kernel_launch(void* const* d_in, const int* in_sizes, int n_in,
                              void* d_out, int out_size, void* d_ws, size_t ws_size,
                              hipStream_t stream) {
  const float* logits  = (const float*)d_in[0];
  const int*   targets = (const int*)d_in[1];
  const int B        = in_sizes[0] / NUM_CLASSES;   // 4,194,304
  const int numTiles = B / ROWS_PER_TILE;           // 16,384 (B divisible by 256)
  float* partials    = (float*)d_ws;                // 2048 floats of scratch

  pom1b_partial<<<NUM_BLOCKS, BLOCK_THREADS, 0, stream>>>(
      logits, targets, partials, numTiles);
  pom1b_finalize<<<1, 256, 0, stream>>>(partials, NUM_BLOCKS, (float*)d_out, B);
}